// ApplyBiasRopeUpdateKVCacheWrapper_59124519796898
// MI455X (gfx1250) — compile-verified
//
#include <hip/hip_runtime.h>
#include <cstddef>

// -------- problem constants (match reference) --------
namespace {
constexpr int kH   = 32;     // query heads
constexpr int kHKV = 8;      // kv heads
constexpr int kD   = 128;    // head dim
constexpr int kB   = 4;      // batch
constexpr int kS   = 4096;   // seq len (power of two)
constexpr int kTPB = 64;     // tokens per cache block
constexpr int kT   = kB * kS;                    // 16384 tokens
constexpr int kRow = (kH + 2 * kHKV) * kD;       // 6144 floats / token
constexpr size_t kCacheElems = (size_t)kT * kHKV * kD;  // 16,777,216 per cache
constexpr float kLn1e4OverHalf = 9.2103403719761836f / 64.0f; // ln(10000)/(D/2)
}

typedef float v2f __attribute__((ext_vector_type(2)));

__device__ __forceinline__ float quant_i8(float x, float scale) {
  // jnp.clip(jnp.round(x*scale), -128, 127) ; round == RTE == rintf (default mode)
  return fminf(fmaxf(rintf(x * scale), -128.0f), 127.0f);
}

__global__ __launch_bounds__(256) void rope_kvcache_kernel(
    const float* __restrict__ qkv,          // [T, 6144] f32
    const int*   __restrict__ block_tables, // [B, S/TPB] i32
    const float* __restrict__ kv_scale,     // [1] f32
    float*       __restrict__ out)          // qkv_out | k_cache | v_cache (flat)
{
  const int tid  = threadIdx.x;
  const int lane = tid & 31;
  const int wave = tid >> 5;

  const float scale = kv_scale[0];

  // Per-lane inverse frequencies for freq indices f0 = 2*lane, f1 = 2*lane+1.
  const float inv0 = __expf(-(float)(2 * lane)     * kLn1e4OverHalf);
  const float inv1 = __expf(-(float)(2 * lane + 1) * kLn1e4OverHalf);

  float* out_qkv = out;
  float* k_cache = out + (size_t)kT * kRow;
  float* v_cache = k_cache + kCacheElems;

  const int a = 2 * lane;       // dims [a, a+1]   (first half)
  const int b = 64 + 2 * lane;  // dims [b, b+1]   (second half)

  for (int t = blockIdx.x; t < kT; t += gridDim.x) {
    const float* row  = qkv     + (size_t)t * kRow;
    float*       orow = out_qkv + (size_t)t * kRow;

    // Prefetch the next token this block will process (global_prefetch_b8).
    const int tn = t + gridDim.x;
    if (tn < kT) {
      const float* nrow = qkv + (size_t)tn * kRow;
      // 8 waves x 32 lanes, 16-float stride per lane -> covers ~4096 of 6144
      // floats at cacheline granularity; the rest rides along in L2 lines.
      __builtin_prefetch(nrow + (wave * (kRow / 8) + lane * 16), 0, 0);
    }

    const int pos   = t & (kS - 1);
    const int batch = t >> 12;                  // t / kS
    const int blk   = block_tables[batch * (kS / kTPB) + (pos >> 6)];
    const int slot  = pos & (kTPB - 1);
    const size_t cbase = ((size_t)blk * kTPB + slot) * (kHKV * kD);

    // Per-token angles for this lane's two frequencies.
    const float p = (float)pos;                 // SCALING == 1
    float s0, c0, s1, c1;
    __sincosf(p * inv0, &s0, &c0);
    __sincosf(p * inv1, &s1, &c1);

    // ---- 4 Q heads: RoPE -> qkv_out only ----
#pragma unroll
    for (int i = 0; i < 4; ++i) {
      const int h = wave + 8 * i;
      const float* hp = row  + h * kD;
      float*       op = orow + h * kD;
      v2f x1 = __builtin_nontemporal_load((const v2f*)(hp + a));
      v2f x2 = __builtin_nontemporal_load((const v2f*)(hp + b));
      v2f o1, o2;
      o1.x = x1.x * c0 - x2.x * s0;
      o1.y = x1.y * c1 - x2.y * s1;
      o2.x = x2.x * c0 + x1.x * s0;
      o2.y = x2.y * c1 + x1.y * s1;
      __builtin_nontemporal_store(o1, (v2f*)(op + a));
      __builtin_nontemporal_store(o2, (v2f*)(op + b));
    }

    // ---- 1 K head per wave: RoPE -> qkv_out + quantize -> k_cache ----
    {
      const int h  = kH + wave;         // 32..39
      const int kh = wave;              // kv-head index
      const float* hp = row  + h * kD;
      float*       op = orow + h * kD;
      v2f x1 = __builtin_nontemporal_load((const v2f*)(hp + a));
      v2f x2 = __builtin_nontemporal_load((const v2f*)(hp + b));
      v2f o1, o2;
      o1.x = x1.x * c0 - x2.x * s0;
      o1.y = x1.y * c1 - x2.y * s1;
      o2.x = x2.x * c0 + x1.x * s0;
      o2.y = x2.y * c1 + x1.y * s1;
      __builtin_nontemporal_store(o1, (v2f*)(op + a));
      __builtin_nontemporal_store(o2, (v2f*)(op + b));

      v2f q1, q2;
      q1.x = quant_i8(o1.x, scale); q1.y = quant_i8(o1.y, scale);
      q2.x = quant_i8(o2.x, scale); q2.y = quant_i8(o2.y, scale);
      float* kc = k_cache + cbase + (size_t)kh * kD;
      __builtin_nontemporal_store(q1, (v2f*)(kc + a));
      __builtin_nontemporal_store(q2, (v2f*)(kc + b));
    }

    // ---- 1 V head per wave: passthrough -> qkv_out + quantize -> v_cache ----
    {
      const int h  = kH + kHKV + wave;  // 40..47
      const int vh = wave;
      const float* hp = row  + h * kD;
      float*       op = orow + h * kD;
      v2f x1 = __builtin_nontemporal_load((const v2f*)(hp + a));
      v2f x2 = __builtin_nontemporal_load((const v2f*)(hp + b));
      __builtin_nontemporal_store(x1, (v2f*)(op + a));
      __builtin_nontemporal_store(x2, (v2f*)(op + b));

      v2f q1, q2;
      q1.x = quant_i8(x1.x, scale); q1.y = quant_i8(x1.y, scale);
      q2.x = quant_i8(x2.x, scale); q2.y = quant_i8(x2.y, scale);
      float* vc = v_cache + cbase + (size_t)vh * kD;
      __builtin_nontemporal_store(q1, (v2f*)(vc + a));
      __builtin_nontemporal_store(q2, (v2f*)(vc + b));
    }
  }
}

extern "C" void kernel_launch(void* const* d_in, const int* in_sizes, int n_in,
                              void* d_out, int out_size, void* d_ws, size_t ws_size,
                              hipStream_t stream) {
  (void)in_sizes; (void)n_in; (void)out_size; (void)d_ws; (void)ws_size;
  const float* qkv          = (const float*)d_in[0];
  const int*   block_tables = (const int*)d_in[1];
  const float* kv_scale     = (const float*)d_in[2];
  float*       out          = (float*)d_out;

  // 2048 blocks x 256 threads (8 waves): grid-stride over 16384 tokens (8/block).
  // Purely bandwidth-bound (~0.94 GB @ 23.3 TB/s ≈ 40 us floor); occupancy +
  // prefetch hide HBM latency, NT hints keep the touch-once streams out of L2.
  dim3 grid(2048), block(256);
  hipLaunchKernelGGL(rope_kvcache_kernel, grid, block, 0, stream,
                     qkv, block_tables, kv_scale, out);
}